// ManifoldConstrainedHyperConnections_31086973288792
// MI455X (gfx1250) — compile-verified
//
#include <hip/hip_runtime.h>

typedef float v2f __attribute__((ext_vector_type(2)));
typedef float v4f __attribute__((ext_vector_type(4)));
typedef float v8f __attribute__((ext_vector_type(8)));

#define D_DIM 1024
#define S_DIM 4
#define EPSV 1e-12f
#define SINK_ITERS 20
#define WAVES_PER_BLOCK 8
#define TOK_PER_WAVE 4   // 4 tokens * 4 streams = 16 rows = WMMA M

__device__ __forceinline__ float sigm(float x) { return 1.0f / (1.0f + __expf(-x)); }

__global__ __launch_bounds__(256) void mchc_fused_kernel(
    const float* __restrict__ residuals,       // (B,N,S,D)
    const float* __restrict__ gamma,           // (D)
    const float* __restrict__ w_alpha,         // (D, S+1)
    const float* __restrict__ static_alpha,    // (S, S+1)
    const float* __restrict__ pre_branch_scale,// scalar
    const float* __restrict__ residual_scale,  // scalar
    const float* __restrict__ w_beta,          // (D)
    const float* __restrict__ static_beta,     // (S)
    const float* __restrict__ h_post_scale,    // scalar
    float* __restrict__ out)                   // (B,N,S,D)
{
    // ---- LDS ----
    __shared__ float sWg[D_DIM * 8];                      // (1+gamma)*[Wa0..Wa4, Wb, 0, 0]  32KB
    __shared__ float sD[WAVES_PER_BLOCK][16][17];         // WMMA result 16x16 (padded)
    __shared__ float sSS[WAVES_PER_BLOCK][16];            // row sum-of-squares
    __shared__ float sApre[WAVES_PER_BLOCK][TOK_PER_WAVE][4];
    __shared__ float sBet [WAVES_PER_BLOCK][TOK_PER_WAVE][4];
    __shared__ float sAres[WAVES_PER_BLOCK][TOK_PER_WAVE][4][4]; // [s'][t]

    const int tid  = threadIdx.x;
    const int lane = tid & 31;
    const int wave = tid >> 5;

    // ---- stage gamma-folded weights into LDS ----
    for (int d = tid; d < D_DIM; d += 256) {
        float g1 = 1.0f + gamma[d];
        float* w = &sWg[d * 8];
#pragma unroll
        for (int m = 0; m < 5; ++m) w[m] = g1 * w_alpha[d * 5 + m];
        w[5] = g1 * w_beta[d];
        w[6] = 0.0f;
        w[7] = 0.0f;
    }
    __syncthreads();

    // ======== Phase 1: WMMA  P = X(16xD) * Wg(Dx16)  +  per-row sumsq ========
    // rows r = 0..15 of this wave's 4-token group: token = r>>2, stream = r&3
    const int rowbase = (blockIdx.x * WAVES_PER_BLOCK + wave) * 16; // global (token*4+stream) row
    const int r    = lane & 15;
    const int koff = (lane >> 4) * 2;           // A layout: half-wave holds K+2,K+3
    const int bcol = lane & 7;                  // B cols 6,7 are zero; cols 8..15 alias (ignored)
    const float* aptr = residuals + (size_t)(rowbase + r) * D_DIM + koff;

    v8f c0 = {}, c1 = {}, c2 = {}, c3 = {};
    float ss = 0.0f;

    for (int k0 = 0; k0 < D_DIM; k0 += 16) {
        v2f a0 = *(const v2f*)(aptr + k0);
        v2f a1 = *(const v2f*)(aptr + k0 + 4);
        v2f a2 = *(const v2f*)(aptr + k0 + 8);
        v2f a3 = *(const v2f*)(aptr + k0 + 12);

        v2f b0 = { sWg[(k0 + koff)      * 8 + bcol], sWg[(k0 + koff + 1)  * 8 + bcol] };
        v2f b1 = { sWg[(k0 + koff + 4)  * 8 + bcol], sWg[(k0 + koff + 5)  * 8 + bcol] };
        v2f b2 = { sWg[(k0 + koff + 8)  * 8 + bcol], sWg[(k0 + koff + 9)  * 8 + bcol] };
        v2f b3 = { sWg[(k0 + koff + 12) * 8 + bcol], sWg[(k0 + koff + 13) * 8 + bcol] };

        ss += a0.x * a0.x + a0.y * a0.y + a1.x * a1.x + a1.y * a1.y
            + a2.x * a2.x + a2.y * a2.y + a3.x * a3.x + a3.y * a3.y;

        c0 = __builtin_amdgcn_wmma_f32_16x16x4_f32(false, a0, false, b0, (short)0, c0, false, false);
        c1 = __builtin_amdgcn_wmma_f32_16x16x4_f32(false, a1, false, b1, (short)0, c1, false, false);
        c2 = __builtin_amdgcn_wmma_f32_16x16x4_f32(false, a2, false, b2, (short)0, c2, false, false);
        c3 = __builtin_amdgcn_wmma_f32_16x16x4_f32(false, a3, false, b3, (short)0, c3, false, false);
    }
    v8f cacc = c0 + c1 + c2 + c3;

    // sumsq: lane l holds partial of row (l&15) for K%4 in {koff,koff+1}; combine half-waves
    ss += __shfl_xor(ss, 16);
    if (lane < 16) sSS[wave][lane] = ss;

    // D layout: VGPR j -> M = j + (lane>=16 ? 8 : 0), N = lane&15
#pragma unroll
    for (int j = 0; j < 8; ++j)
        sD[wave][j + ((lane >> 4) * 8)][lane & 15] = cacc[j];
    __syncthreads();

    // ======== Phase 2: per-token small math + lane-parallel Sinkhorn ========
    const float preS = pre_branch_scale[0];
    const float resS = residual_scale[0];
    const float hps  = h_post_scale[0];

    for (int p = 0; p < 2; ++p) {                 // 2 tokens per round, lanes 0-15 / 16-31
        const int ltok = 2 * p + (lane >> 4);
        const int s    = (lane >> 2) & 3;
        const int t    = lane & 3;
        const int M    = ltok * 4 + s;

        const float inv = 32.0f / fmaxf(sqrtf(sSS[wave][M]), EPSV); // sqrt(D)=32
        // log_alpha entry for sinkhorn (columns 1..4 of alpha)
        float aval = sD[wave][M][t + 1] * inv * resS + static_alpha[s * 5 + t + 1];
        // subtract col max (max over s: lane bits 2..3)
        float mx = fmaxf(aval, __shfl_xor(aval, 4));
        mx = fmaxf(mx, __shfl_xor(mx, 8));
        float a = __expf(aval - mx);
#pragma unroll
        for (int it = 0; it < SINK_ITERS; ++it) {
            float cs = a + __shfl_xor(a, 4);  cs += __shfl_xor(cs, 8);   // sum over s
            a = a / fmaxf(cs, EPSV);
            float rs = a + __shfl_xor(a, 1);  rs += __shfl_xor(rs, 2);   // sum over t
            a = a / fmaxf(rs, EPSV);
        }
        sAres[wave][ltok][s][t] = a;
        if (t == 0)
            sApre[wave][ltok][s] = sigm(sD[wave][M][0] * inv * preS + static_alpha[s * 5]);
        if (t == 1)
            sBet[wave][ltok][s] = 2.0f * sigm(sD[wave][M][5] * inv * hps + static_beta[s]);
    }
    __syncthreads();

    // ======== Phase 3: mix + beta-gated output (x re-read hits L2) ========
    for (int ltok = 0; ltok < TOK_PER_WAVE; ++ltok) {
        float ap[4], bt[4], ar[4][4];
#pragma unroll
        for (int s = 0; s < 4; ++s) {
            ap[s] = sApre[wave][ltok][s];
            bt[s] = sBet[wave][ltok][s];
#pragma unroll
            for (int t = 0; t < 4; ++t) ar[s][t] = sAres[wave][ltok][s][t];
        }
        const size_t tb = (size_t)(rowbase + ltok * 4) * D_DIM;
#pragma unroll
        for (int k = 0; k < 8; ++k) {
            const int cidx = k * 128 + lane * 4;
            v4f x0 = *(const v4f*)(residuals + tb + 0 * D_DIM + cidx);
            v4f x1 = *(const v4f*)(residuals + tb + 1 * D_DIM + cidx);
            v4f x2 = *(const v4f*)(residuals + tb + 2 * D_DIM + cidx);
            v4f x3 = *(const v4f*)(residuals + tb + 3 * D_DIM + cidx);
            v4f branch = ap[0] * x0 + ap[1] * x1 + ap[2] * x2 + ap[3] * x3;
#pragma unroll
            for (int t = 0; t < 4; ++t) {
                v4f o = branch * bt[t]
                      + ar[0][t] * x0 + ar[1][t] * x1 + ar[2][t] * x2 + ar[3][t] * x3;
                *(v4f*)(out + tb + (size_t)t * D_DIM + cidx) = o;
            }
        }
    }
}

extern "C" void kernel_launch(void* const* d_in, const int* in_sizes, int n_in,
                              void* d_out, int out_size, void* d_ws, size_t ws_size,
                              hipStream_t stream) {
    const float* residuals = (const float*)d_in[0];
    const float* gamma     = (const float*)d_in[1];
    const float* w_alpha   = (const float*)d_in[2];
    const float* st_alpha  = (const float*)d_in[3];
    const float* pre_s     = (const float*)d_in[4];
    const float* res_s     = (const float*)d_in[5];
    const float* w_beta    = (const float*)d_in[6];
    const float* st_beta   = (const float*)d_in[7];
    const float* hps       = (const float*)d_in[8];
    float* out = (float*)d_out;

    const int tokens = in_sizes[0] / (S_DIM * D_DIM);        // B*N = 16384
    const int blocks = tokens / (WAVES_PER_BLOCK * TOK_PER_WAVE); // 512

    mchc_fused_kernel<<<blocks, 256, 0, stream>>>(
        residuals, gamma, w_alpha, st_alpha, pre_s, res_s, w_beta, st_beta, hps, out);
}